// PEPS_NNproj_Model_3178275799240
// MI455X (gfx1250) — compile-verified
//
#include <hip/hip_runtime.h>
#include <hip/hip_bf16.h>
#include <math.h>

typedef __attribute__((ext_vector_type(2))) float v2f;
typedef __attribute__((ext_vector_type(8))) float v8f;

__device__ __forceinline__ v8f wmma4(v2f a, v2f b, v8f c) {
  // D = A(16x4) * B(4x16) + C, all f32  -> v_wmma_f32_16x16x4_f32
  return __builtin_amdgcn_wmma_f32_16x16x4_f32(false, a, false, b, (short)0, c, false, false);
}

__device__ __forceinline__ float wsum32(float v) {
#pragma unroll
  for (int off = 16; off > 0; off >>= 1) v += __shfl_xor(v, off, 32);
  return v;
}

// C[M x N] = A * B with generic strides: A(i,k)=A[i*sar+k*sac], B(k,j)=B[k*sbr+j*sbc].
// Executed by one full wave32. Guards handle M,N,K not multiples of the 16/4 tile.
__device__ void wmma_gemm(const float* A, int sar, int sac,
                          const float* B, int sbr, int sbc,
                          float* C, int ldc, int M, int N, int K)
{
  const int lane = threadIdx.x;
  const int li = lane & 15;
  const int kh = (lane >> 4) << 1;          // A/B K-pair base per lane half
  for (int mb = 0; mb < M; mb += 16) {
    const int mi = mb + li;
    for (int nb = 0; nb < N; nb += 16) {
      const int ni = nb + li;
      v8f acc = {0.f,0.f,0.f,0.f,0.f,0.f,0.f,0.f};
      for (int k0 = 0; k0 < K; k0 += 4) {
        const int ka = k0 + kh;
        v2f a, b;
        a.x = (mi < M && ka     < K) ? A[mi*sar + (ka    )*sac] : 0.f;
        a.y = (mi < M && (ka+1) < K) ? A[mi*sar + (ka + 1)*sac] : 0.f;
        b.x = (ni < N && ka     < K) ? B[(ka    )*sbr + ni*sbc] : 0.f;
        b.y = (ni < N && (ka+1) < K) ? B[(ka + 1)*sbr + ni*sbc] : 0.f;
        acc = wmma4(a, b, acc);
      }
      const int cn = nb + li;
      const int mbase = mb + ((lane >> 4) << 3);
      if (cn < N) {
#pragma unroll
        for (int r = 0; r < 8; ++r) {
          const int cm = mbase + r;
          if (cm < M) C[cm*ldc + cn] = acc[r];
        }
      }
    }
  }
}

// In-place upper Cholesky factor R of SPD G (m x m): G = R^T R; lower cleared.
__device__ void chol_upper(float* G, int m) {
  const int lane = threadIdx.x;
  for (int k = 0; k < m; ++k) {
    __syncthreads();
    if (lane == 0) G[k*m+k] = sqrtf(fmaxf(G[k*m+k], 1e-20f));
    __syncthreads();
    const float inv = 1.0f / G[k*m+k];
    for (int j = k + 1 + lane; j < m; j += 32) G[k*m+j] *= inv;
    __syncthreads();
    const int rem = m - k - 1;
    for (int e = lane; e < rem*rem; e += 32) {
      const int i = k + 1 + e / rem;
      const int j = k + 1 + e % rem;
      if (j >= i) G[i*m+j] -= G[k*m+i] * G[k*m+j];
    }
  }
  __syncthreads();
  for (int e = lane; e < m*m; e += 32)
    if ((e % m) < (e / m)) G[e] = 0.f;
  __syncthreads();
}

// T layout: (Lx=6, Ly=6, D=4, D, D, D, PHYS=2) row-major
__device__ __forceinline__ int tidx(int i, int j, int a, int R, int b, int x, int p) {
  return (((((i*6 + j)*4 + a)*4 + R)*4 + b)*4 + x)*2 + p;
}

// ---------------- MLP kernels (batched WMMA GEMMs) ----------------

__global__ __launch_bounds__(32)
void mlp1_kernel(const int* __restrict__ x, const float* __restrict__ W1,
                 const float* __restrict__ b1, float* __restrict__ H)
{
  const int lane = threadIdx.x;
  const int tm = blockIdx.x >> 2;   // 64 row tiles of 1024
  const int tn = blockIdx.x & 3;    // 4 col tiles of 64
  const int li = lane & 15;
  const int kh = (lane >> 4) << 1;
  const int mi = tm*16 + li;
  const int ni = tn*16 + li;
  v8f acc = {0.f,0.f,0.f,0.f,0.f,0.f,0.f,0.f};
  for (int k0 = 0; k0 < 36; k0 += 4) {      // K = NSITES = 36
    const int ka = k0 + kh;
    v2f a, b;
    a.x = (float)x[mi*36 + ka];
    a.y = (float)x[mi*36 + ka + 1];
    b.x = W1[ni*36 + ka];
    b.y = W1[ni*36 + ka + 1];
    acc = wmma4(a, b, acc);
  }
  const int cn = tn*16 + li;
  const int mbase = tm*16 + ((lane >> 4) << 3);
  const float bias = b1[cn];
#pragma unroll
  for (int r = 0; r < 8; ++r)
    H[(mbase + r)*64 + cn] = fmaxf(acc[r] + bias, 0.f);
}

__global__ __launch_bounds__(32)
void mlp2_kernel(const float* __restrict__ H, const float* __restrict__ W2,
                 const float* __restrict__ b2, float* __restrict__ delta)
{
  const int lane = threadIdx.x;
  const int tm = blockIdx.x / 400;  // 64 row tiles
  const int tn = blockIdx.x % 400;  // 400 col tiles of 6400
  const int li = lane & 15;
  const int kh = (lane >> 4) << 1;
  const int mi = tm*16 + li;
  const int ni = tn*16 + li;
  v8f acc = {0.f,0.f,0.f,0.f,0.f,0.f,0.f,0.f};
  for (int k0 = 0; k0 < 64; k0 += 4) {      // K = NHID = 64
    const int ka = k0 + kh;
    v2f a, b;
    a.x = H[mi*64 + ka];
    a.y = H[mi*64 + ka + 1];
    b.x = W2[ni*64 + ka];
    b.y = W2[ni*64 + ka + 1];
    acc = wmma4(a, b, acc);
  }
  const int cn = tn*16 + li;
  const int mbase = tm*16 + ((lane >> 4) << 3);
  const float bias = b2[cn];
#pragma unroll
  for (int r = 0; r < 8; ++r)
    delta[(size_t)(mbase + r)*6400 + cn] = 0.001f * (acc[r] + bias);  // ETA pre-applied
}

// ---------------- PEPS sweep kernel: one sample per wave32 ----------------

__global__ __launch_bounds__(32)
void peps_kernel(const float* __restrict__ Tg, const int* __restrict__ xall,
                 const float* __restrict__ delta, float* __restrict__ plist,
                 float* __restrict__ bscratch, float* __restrict__ out, int nsamp)
{
  __shared__ float SH[10240];
  __shared__ int   OrdI[16];
  __shared__ float Svs[48];              // [0..31] sigma, [32..47] 1/sqrt(s+eps)
  float* G1s = SH;                       // 1024: Gram1 -> R1
  float* G2s = SH + 1024;               // 1024: Gram2 -> R2
  float* Mms = SH + 2048;               // 1024: R1*R2^T -> U
  float* Vvs = SH + 3072;               // 1024: V
  float* PdS = SH + 4096;               // 512: Pd padded m x 16
  float* PuS = SH + 4608;               // 512: Pu padded m x 16
  float* TMP = SH + 5120;               // 2048: truncation temporaries
  float* BIN = SH + 7168;               // 6*256: entering boundary tensors
  float* VEC = SH + 8704;               // 64: chain vectors
  const int lane = threadIdx.x;
  const int Arow[6] = {1,4,4,4,4,4};
  const int Brow[6] = {4,4,4,4,4,1};
  const int cbt[5]  = {4,8,8,8,4};      // static truncation ranks per pair
  float* myb = bscratch + (size_t)blockIdx.x * 24576;  // 6 x 4096 floats

  for (int s = blockIdx.x; s < nsamp; s += gridDim.x) {
    const int* xg = xall + s*36;
    float amp = 0.f;
    for (int pass = 0; pass < 2; ++pass) {
      int uin[6], din[6];
      // ---- row 0 init: B[i] = A[i][0][:,:,:,0]
      for (int i = 0; i < 6; ++i) {
        const int A_ = Arow[i], B_ = Brow[i];
        const int phys = xg[i*6 + 0];
        const int cnt = A_*4*B_;
        for (int e = lane; e < cnt; e += 32) {
          const int b = e % B_;
          const int R = (e / B_) & 3;
          const int a = e / (B_*4);
          BIN[i*256 + e] = Tg[tidx(i,0,a,R,b,0,phys)];
        }
        uin[i] = A_; din[i] = B_;
      }
      __syncthreads();
      int poff = 0;
      for (int j = 1; j <= 4; ++j) {
        int F[6], Lc[6];
        // ---- absorb row j:  nt[u,a,R,d,b] = sum_x B[u,x,d] * t[a,R,b,x]
        for (int i = 0; i < 6; ++i) {
          const int A_ = Arow[i], B_ = Brow[i];
          const int u_ = uin[i],  d_ = din[i];
          const int Dd_ = d_*B_;
          const int phys = xg[i*6 + j];
          float* Bn = myb + i*4096;
          const int cnt = u_*A_*4*Dd_;
          for (int e = lane; e < cnt; e += 32) {
            const int col = e % Dd_;
            const int dd = col / B_, b = col % B_;
            const int row2 = e / Dd_;
            const int R = row2 & 3;
            const int ua = row2 >> 2;
            const int u = ua / A_, a = ua % A_;
            float acc = 0.f;
#pragma unroll
            for (int xx = 0; xx < 4; ++xx)
              acc += BIN[i*256 + (u*4 + xx)*d_ + dd] * Tg[tidx(i,j,a,R,b,xx,phys)];
            Bn[e] = acc;
          }
          F[i] = u_*A_;  Lc[i] = Dd_;
        }
        __syncthreads();
        // ---- 5 bond pairs
        for (int i = 0; i < 5; ++i) {
          const int m = Lc[i];             // 16 or 32
          const int rowsT = F[i]*4;        // 4..32
          const int L = 4*Lc[i+1];         // 4..128
          const int cb = cbt[i];
          float* mt = myb + i*4096;        // (rowsT x m)
          float* mb = myb + (i+1)*4096;    // (m x L)
          if (pass == 0) {
            // Gram matrices via WMMA, Cholesky -> R factors
            wmma_gemm(mt, 1, m, mt, m, 1, G1s, m, m, m, rowsT);   // G1 = mt^T mt
            __syncthreads();
            wmma_gemm(mb, L, 1, mb, 1, L, G2s, m, m, m, L);       // G2 = mb mb^T
            __syncthreads();
            chol_upper(G1s, m);
            chol_upper(G2s, m);
            wmma_gemm(G1s, m, 1, G2s, 1, m, Mms, m, m, m, m);     // M = R1 R2^T
            __syncthreads();
            for (int e = lane; e < m*m; e += 32) Vvs[e] = ((e/m) == (e%m)) ? 1.f : 0.f;
            __syncthreads();
            // one-sided Jacobi SVD: lane r owns row r; dots via wave32 shfl reductions
            for (int sw = 0; sw < 8; ++sw)
              for (int p = 0; p < m-1; ++p)
                for (int q = p+1; q < m; ++q) {
                  float mp = (lane < m) ? Mms[lane*m+p] : 0.f;
                  float mq = (lane < m) ? Mms[lane*m+q] : 0.f;
                  const float app = wsum32(mp*mp);
                  const float aqq = wsum32(mq*mq);
                  const float apq = wsum32(mp*mq);
                  if (fabsf(apq) > 1e-12f*sqrtf(app*aqq) + 1e-30f) {
                    const float tau = (aqq - app) / (2.f*apq);
                    const float t = ((tau >= 0.f) ? 1.f : -1.f) /
                                    (fabsf(tau) + sqrtf(1.f + tau*tau));
                    const float c = rsqrtf(1.f + t*t);
                    const float sn = t*c;
                    if (lane < m) {
                      Mms[lane*m+p] = c*mp - sn*mq;
                      Mms[lane*m+q] = sn*mp + c*mq;
                      const float vp = Vvs[lane*m+p], vq = Vvs[lane*m+q];
                      Vvs[lane*m+p] = c*vp - sn*vq;
                      Vvs[lane*m+q] = sn*vp + c*vq;
                    }
                  }
                }
            __syncthreads();
            for (int q = 0; q < m; ++q) {
              const float v = (lane < m) ? Mms[lane*m+q] : 0.f;
              const float nn = wsum32(v*v);
              if (lane == 0) Svs[q] = sqrtf(nn);
            }
            __syncthreads();
            for (int e = lane; e < m*m; e += 32) Mms[e] /= fmaxf(Svs[e % m], 1e-30f);
            if (lane == 0) {                       // top-cb selection, descending
              unsigned used = 0;
              for (int c = 0; c < cb; ++c) {
                int best = 0; float bv = -1.f;
                for (int q = 0; q < m; ++q)
                  if (!((used >> q) & 1u) && Svs[q] > bv) { bv = Svs[q]; best = q; }
                used |= (1u << best);
                OrdI[c] = best;
                Svs[32 + c] = rsqrtf(bv + 1e-12f);
              }
            }
            __syncthreads();
            // Pd = R2^T V_sel * sinv ; Pu = R1^T U_sel * sinv  (padded m x 16)
            for (int e = lane; e < m*16; e += 32) {
              const int r = e >> 4, c = e & 15;
              float pd = 0.f, pu = 0.f;
              if (c < cb) {
                const int oc = OrdI[c];
                float a1 = 0.f, a2 = 0.f;
                for (int k = 0; k < m; ++k) {
                  a1 += G2s[k*m + r] * Vvs[k*m + oc];
                  a2 += G1s[k*m + r] * Mms[k*m + oc];
                }
                const float si = Svs[32 + c];
                pd = a1 * si;  pu = a2 * si;
              }
              PdS[e] = pd;  PuS[e] = pu;
            }
            __syncthreads();
            float* pl = plist + (size_t)s*6400 + poff;   // store (m x cb) row-major
            for (int e = lane; e < m*cb; e += 32) {
              const int r = e / cb, c = e % cb;
              pl[e]        = PdS[r*16 + c];
              pl[m*cb + e] = PuS[r*16 + c];
            }
          } else {
            // sweep 2: perturbed projectors = plist + eta*mlp (delta pre-scaled)
            const float* pl = plist + (size_t)s*6400 + poff;
            const float* dl = delta + (size_t)s*6400 + poff;
            for (int e = lane; e < m*16; e += 32) {
              const int r = e >> 4, c = e & 15;
              float pd = 0.f, pu = 0.f;
              if (c < cb) {
                pd = pl[r*cb + c]        + dl[r*cb + c];
                pu = pl[m*cb + r*cb + c] + dl[m*cb + r*cb + c];
              }
              PdS[e] = pd;  PuS[e] = pu;
            }
            __syncthreads();
          }
          // truncate top:  (rowsT x m)(m x 16) -> TMP, keep cb cols
          wmma_gemm(mt, m, 1, PdS, 16, 1, TMP, 16, rowsT, 16, m);
          __syncthreads();
          for (int e = lane; e < rowsT*cb; e += 32)
            mt[e] = TMP[(e / cb)*16 + (e % cb)];
          __syncthreads();
          // truncate bottom: Pu^T (cb x m) * mb (m x L)
          wmma_gemm(PuS, 1, 16, mb, L, 1, TMP, L, cb, L, m);
          __syncthreads();
          for (int e = lane; e < cb*L; e += 32) mb[e] = TMP[e];
          __syncthreads();
          Lc[i] = cb;  F[i+1] = cb;
          poff += 2*m*cb;
        }
        // ---- row exit: compacted Bn -> LDS entering tensors
        for (int i = 0; i < 6; ++i) {
          const int cnt = F[i]*4*Lc[i];
          float* Bn = myb + i*4096;
          for (int e = lane; e < cnt; e += 32) BIN[i*256 + e] = Bn[e];
          uin[i] = F[i];  din[i] = Lc[i];
        }
        __syncthreads();
      }
      if (pass == 1) {
        // ---- finalize: Cs[i] = reshape(einsum('uxd,abx->uadb', B[i], t)), chain product
        float* Cs = SH;  // reuse factor-matrix region (6 x 1024)
        for (int i = 0; i < 6; ++i) {
          const int A_ = Arow[i], B_ = Brow[i];
          const int u_ = uin[i],  d_ = din[i];
          const int rowsC = u_*A_, colsC = d_*B_;
          const int phys = xg[i*6 + 5];
          const int cnt = rowsC*colsC;
          for (int e = lane; e < cnt; e += 32) {
            const int col = e % colsC;
            const int dd = col / B_, b = col % B_;
            const int ra = e / colsC;
            const int u = ra / A_, a = ra % A_;
            float acc = 0.f;
#pragma unroll
            for (int xx = 0; xx < 4; ++xx)
              acc += BIN[i*256 + (u*4 + xx)*d_ + dd] * Tg[tidx(i,5,a,0,b,xx,phys)];
            Cs[i*1024 + e] = acc;
          }
        }
        __syncthreads();
        float* cur = VEC;
        float* nxt = VEC + 32;
        const int len0 = din[0]*Brow[0];
        if (lane < len0) cur[lane] = Cs[lane];     // Cs[0] is (1 x len0)
        __syncthreads();
        for (int i = 1; i < 6; ++i) {
          const int rowsC = uin[i]*Arow[i];
          const int colsC = din[i]*Brow[i];
          if (lane < colsC) {
            float acc = 0.f;
            for (int p = 0; p < rowsC; ++p) acc += cur[p] * Cs[i*1024 + p*colsC + lane];
            nxt[lane] = acc;
          }
          __syncthreads();
          float* tsw = cur; cur = nxt; nxt = tsw;
        }
        amp = cur[0];
        __syncthreads();
      }
    }
    if (lane == 0) out[s] = amp;
    __syncthreads();
  }
}

extern "C" void kernel_launch(void* const* d_in, const int* in_sizes, int n_in,
                              void* d_out, int out_size, void* d_ws, size_t ws_size,
                              hipStream_t stream) {
  (void)in_sizes; (void)n_in; (void)out_size;
  const float* T  = (const float*)d_in[0];
  const float* W1 = (const float*)d_in[1];
  const float* b1 = (const float*)d_in[2];
  const float* W2 = (const float*)d_in[3];
  const float* b2 = (const float*)d_in[4];
  const int*   x  = (const int*)d_in[5];
  float* out = (float*)d_out;
  float* ws = (float*)d_ws;
  float* Hbuf  = ws;                       // 1024*64
  float* delta = Hbuf + 65536;             // 1024*6400
  float* plist = delta + 6553600;          // 1024*6400
  float* bsc   = plist + 6553600;          // nblk * 24576
  const size_t wsf = ws_size / sizeof(float);
  const size_t fixedf = 65536ull + 2ull*6553600ull;
  int nblk = 1;
  if (wsf > fixedf + 24576ull) {
    size_t nb = (wsf - fixedf) / 24576ull;
    if (nb > 512) nb = 512;
    nblk = (int)nb;
  }
  mlp1_kernel<<<256, 32, 0, stream>>>(x, W1, b1, Hbuf);
  mlp2_kernel<<<25600, 32, 0, stream>>>(Hbuf, W2, b2, delta);
  peps_kernel<<<nblk, 32, 0, stream>>>(T, x, delta, plist, bsc, out, 1024);
}